// SSD_50895362458279
// MI455X (gfx1250) — compile-verified
//
#include <hip/hip_runtime.h>
#include <hip/hip_bf16.h>
#include <stdint.h>

// ---------------------------------------------------------------------------
// SSD forward for MI455X (gfx1250, wave32).
// Convs: implicit GEMM on V_WMMA_F32_16X16X4_F32 (fp32 matrix path).
// K ordered (tap, ci): no div/mod in the hot loop.
// All hot loads are GLOBAL saddr+voffset (scale_offset) form; branchless
// clamped addressing + cndmask; 4 pixel tiles per wave share each A operand.
// ---------------------------------------------------------------------------

typedef __attribute__((ext_vector_type(2))) float v2f;
typedef __attribute__((ext_vector_type(8))) float v8f;

// force GLOBAL (addrspace 1) memory ops -- avoids FLAT lowering
typedef const float __attribute__((address_space(1)))* gcfp;
typedef       float __attribute__((address_space(1)))* gfp;
__device__ __forceinline__ float ldg(const float* p) { return *(gcfp)p; }
__device__ __forceinline__ void  stg(float* p, float v) { *(gfp)p = v; }

#define BATCH  8
#define NCLASS 20
#define NANCH  24528   // total anchors over 6 scales
#define TN     4       // pixel tiles (of 16) per wave

// ---- per-scale constant tables (match reference _make_anchors) ----
__constant__ int   g_aoff[7]   = {0, 16384, 22528, 24064, 24448, 24512, 24528};
__constant__ int   g_f[6]      = {64, 32, 16, 8, 4, 2};
__constant__ float g_step[6]   = {8.f, 16.f, 32.f, 64.f, 128.f, 256.f};
__constant__ float g_s1[6]     = {51.2f, 102.4f, 189.4f, 276.4f, 363.5f, 450.6f};
__constant__ float g_s2[6]     = {102.4f, 189.4f, 276.4f, 363.5f, 450.6f, 537.6f};
__constant__ int   g_na[6]     = {4, 6, 6, 6, 4, 4};
__constant__ float g_rex[6][4] = {
    {2.f, 0.5f, 0.f, 0.f},
    {2.f, 0.5f, 3.f, 0.33333334f},
    {2.f, 0.5f, 3.f, 0.33333334f},
    {2.f, 0.5f, 3.f, 0.33333334f},
    {2.f, 0.5f, 0.f, 0.f},
    {2.f, 0.5f, 0.f, 0.f}};
// head output offsets (floats, relative to ws base; alias A0 region)
__constant__ long long g_clsoff[6] = {0LL, 2752512LL, 3784704LL, 4042752LL, 4107264LL, 4118016LL};
__constant__ long long g_boxoff[6] = {4120704LL, 4644992LL, 4841600LL, 4890752LL, 4903040LL, 4905088LL};

// ---------------------------------------------------------------------------
// Generic 3x3 conv (stride 1 or 2, 'SAME'), implicit GEMM with f32 WMMA.
// One wave computes a 16(Cout) x 64(pixel) tile: 4 accumulators share each A.
// K = 9*Cin ordered (tap, ci); groups of 4 ci per WMMA.
// ---------------------------------------------------------------------------
__global__ __launch_bounds__(32) void conv3x3_wmma_f32(
    const float* __restrict__ in, const float* __restrict__ w,
    const float* __restrict__ bias, float* __restrict__ out,
    int Cin, int Hin, int Win, int Cout, int Hout, int Wout,
    int stride, int do_relu)
{
    const int lane  = threadIdx.x;       // 0..31
    const int col   = lane & 15;
    const int khalf = lane >> 4;         // upper lane half holds K+2,K+3

    const int pt = blockIdx.x;           // 64-pixel tile
    const int mt = blockIdx.y;           // cout tile
    const int n  = blockIdx.z;           // batch

    const int P   = Hout * Wout;
    const int pl  = (stride == 1) ? 1 : 0;  // SAME pad-lo (stride2 even dims: 0)
    const int HWi = Hin * Win;              // all tensors < 2^31 elements

    // per-tile output pixel for this lane (B-matrix column of tile t)
    int  hbase[TN], wbase[TN], hoT[TN], woT[TN];
    bool pval[TN];
    #pragma unroll
    for (int t = 0; t < TN; ++t) {
        int p   = pt * (16 * TN) + t * 16 + col;
        pval[t] = (p < P);
        int pc  = pval[t] ? p : 0;               // clamp: garbage cols never stored
        int wo  = pc % Wout;
        int ho  = pc / Wout;
        hoT[t] = ho; woT[t] = wo;
        hbase[t] = ho * stride - pl;
        wbase[t] = wo * stride - pl;
    }

    // A-matrix row for this lane (clamped: garbage rows never stored)
    int mrow = mt * 16 + col;
    if (mrow >= Cout) mrow = 0;

    const float* inb = in + (size_t)n * Cin * HWi;   // uniform base (saddr)

    v8f zero = {};
    v8f acc[TN];
    #pragma unroll
    for (int t = 0; t < TN; ++t) acc[t] = zero;

    if ((Cin & 3) == 0) {
        // ------- main path: Cin multiple of 4 (all layers except layer 0) ----
        for (int tap = 0; tap < 9; ++tap) {
            const int kh = tap / 3;
            const int kw = tap - kh * 3;

            // per-tile clamped offsets + padding-validity (loop-invariant in ci)
            int  ipo0[TN], ipo1[TN];
            bool bval[TN];
            #pragma unroll
            for (int t = 0; t < TN; ++t) {
                int hi = hbase[t] + kh;
                int wi = wbase[t] + kw;
                bool v = ((unsigned)hi < (unsigned)Hin) && ((unsigned)wi < (unsigned)Win);
                bval[t] = v;
                int hic = v ? hi : 0;            // clamp to a valid address
                int wic = v ? wi : 0;
                ipo0[t] = hic * Win + wic + (khalf * 2) * HWi;  // ci = khalf*2
                ipo1[t] = ipo0[t] + HWi;                        // ci = khalf*2+1
            }
            int wpo = (mrow * Cin + khalf * 2) * 9 + tap;

            #pragma unroll 2
            for (int ci0 = 0; ci0 < Cin; ci0 += 4) {
                v2f A;
                A.x = ldg(w + wpo);
                A.y = ldg(w + wpo + 9);
                #pragma unroll
                for (int t = 0; t < TN; ++t) {
                    float r0 = ldg(inb + ipo0[t]);   // unconditional, clamped addr
                    float r1 = ldg(inb + ipo1[t]);
                    v2f B;
                    B.x = bval[t] ? r0 : 0.f;        // cndmask, no exec branches
                    B.y = bval[t] ? r1 : 0.f;
                    acc[t] = __builtin_amdgcn_wmma_f32_16x16x4_f32(
                        false, A, false, B, (short)0, acc[t], false, false);
                }
                wpo += 36;                           // 4 ci * 9 floats
                #pragma unroll
                for (int t = 0; t < TN; ++t) { ipo0[t] += 4 * HWi; ipo1[t] += 4 * HWi; }
            }
        }
    } else {
        // ------- tail path: layer 0 (Cin == 3): one padded K-group per tap ---
        int  ci = khalf * 2;
        bool c0 = (ci     < Cin);
        bool c1 = (ci + 1 < Cin);
        for (int tap = 0; tap < 9; ++tap) {
            const int kh = tap / 3;
            const int kw = tap - kh * 3;
            int wpo = (mrow * Cin + ci) * 9 + tap;
            float a0 = ldg(w + wpo);                 // in range when c0
            float a1 = ldg(w + wpo + (c1 ? 9 : 0));  // clamped address
            v2f A;
            A.x = c0 ? a0 : 0.f;
            A.y = c1 ? a1 : 0.f;
            #pragma unroll
            for (int t = 0; t < TN; ++t) {
                int hi = hbase[t] + kh;
                int wi = wbase[t] + kw;
                bool v = ((unsigned)hi < (unsigned)Hin) && ((unsigned)wi < (unsigned)Win);
                int hic = v ? hi : 0;
                int wic = v ? wi : 0;
                int o0  = hic * Win + wic + ci * HWi;
                float r0 = ldg(inb + o0);
                float r1 = ldg(inb + o0 + (c1 ? HWi : 0));
                v2f B;
                B.x = (v && c0) ? r0 : 0.f;
                B.y = (v && c1) ? r1 : 0.f;
                acc[t] = __builtin_amdgcn_wmma_f32_16x16x4_f32(
                    false, A, false, B, (short)0, acc[t], false, false);
            }
        }
    }

    // fused bias + relu + store.  D layout: VGPR r -> (M = r + khalf*8, N = col)
    float bv[8];
    #pragma unroll
    for (int r = 0; r < 8; ++r) {
        int mm = mt * 16 + r + khalf * 8;
        bv[r] = (bias && mm < Cout) ? ldg(bias + mm) : 0.f;
    }
    float* outb = out + (size_t)n * Cout * P;
    #pragma unroll
    for (int t = 0; t < TN; ++t) {
        if (!pval[t]) continue;
        int pix = hoT[t] * Wout + woT[t];
        #pragma unroll
        for (int r = 0; r < 8; ++r) {
            int mm = mt * 16 + r + khalf * 8;
            if (mm < Cout) {
                float v = acc[t][r] + bv[r];
                if (do_relu) v = fmaxf(v, 0.f);
                stg(outb + mm * P + pix, v);
            }
        }
    }
}

// ---------------------------------------------------------------------------
// Post-processing: softmax over 21 classes, anchor synthesis, box decode,
// nontemporal scatter into (ids | scores | bboxes).  1 thread per (b, anchor).
// ---------------------------------------------------------------------------
__global__ __launch_bounds__(256) void ssd_post(
    const float* __restrict__ ws,
    float* __restrict__ ids, float* __restrict__ sc, float* __restrict__ bb)
{
    int tid = blockIdx.x * blockDim.x + threadIdx.x;
    if (tid >= BATCH * NANCH) return;
    int b = tid / NANCH;
    int n = tid - b * NANCH;

    // which scale
    int s = 0;
    #pragma unroll
    for (int i = 0; i < 5; ++i) s += (n >= g_aoff[i + 1]) ? 1 : 0;
    int rel  = n - g_aoff[s];
    int na   = g_na[s];
    int f    = g_f[s];
    int a    = rel % na;
    int cell = rel / na;
    int x    = cell % f;
    int y    = cell / f;
    int fs   = f * f;

    // ---- softmax over 21 logits (channel-strided gather) ----
    const float* cp = ws + g_clsoff[s]
                    + ((long long)(b * (na * 21) + a * 21) * fs) + (long long)y * f + x;
    float lg[21];
    float mx = -1e30f;
    #pragma unroll
    for (int c = 0; c < 21; ++c) { lg[c] = ldg(cp + (long long)c * fs); mx = fmaxf(mx, lg[c]); }
    float sum = 0.f;
    #pragma unroll
    for (int c = 0; c < 21; ++c) { lg[c] = __expf(lg[c] - mx); sum += lg[c]; }
    float inv = 1.f / sum;

    // ---- box decode ----
    const float* bp = ws + g_boxoff[s]
                    + ((long long)(b * (na * 4) + a * 4) * fs) + (long long)y * f + x;
    float t0 = ldg(bp), t1 = ldg(bp + fs);
    float t2 = ldg(bp + 2 * (long long)fs), t3 = ldg(bp + 3 * (long long)fs);

    float step = g_step[s], s1 = g_s1[s];
    float cx = (x + 0.5f) * step, cy = (y + 0.5f) * step;
    float aw, ah;
    if (a == 0)      { aw = ah = s1; }
    else if (a == 1) { aw = ah = sqrtf(s1 * g_s2[s]); }
    else             { float q = sqrtf(g_rex[s][a - 2]); aw = s1 * q; ah = s1 / q; }

    float px = t0 * 0.1f * aw + cx;
    float py = t1 * 0.1f * ah + cy;
    float hw = expf(t2 * 0.2f) * aw * 0.5f;
    float hh = expf(t3 * 0.2f) * ah * 0.5f;
    float b0 = px - hw, b1 = py - hh, b2 = px + hw, b3 = py + hh;

    // ---- scatter per class: write-once streaming -> nontemporal stores ----
    #pragma unroll 4
    for (int c = 0; c < NCLASS; ++c) {
        float pr = lg[c + 1] * inv;
        bool msk = pr > 0.01f;
        long long idx = ((long long)b * NCLASS + c) * NANCH + n;
        __builtin_nontemporal_store(msk ? (float)c : -1.f, ids + idx);
        __builtin_nontemporal_store(msk ? pr : 0.f,        sc + idx);
        float* o = bb + idx * 4;
        __builtin_nontemporal_store(b0, o + 0);
        __builtin_nontemporal_store(b1, o + 1);
        __builtin_nontemporal_store(b2, o + 2);
        __builtin_nontemporal_store(b3, o + 3);
    }
}

// ---------------------------------------------------------------------------
static inline void launch_conv(const float* in, const float* w, const float* bias,
                               float* out, int Cin, int Hin, int Win,
                               int Cout, int Hout, int Wout, int stride, int relu,
                               hipStream_t stream)
{
    dim3 grid((Hout * Wout + (16 * TN - 1)) / (16 * TN), (Cout + 15) / 16, BATCH);
    conv3x3_wmma_f32<<<grid, 32, 0, stream>>>(in, w, bias, out,
                                              Cin, Hin, Win, Cout, Hout, Wout,
                                              stride, relu);
}

extern "C" void kernel_launch(void* const* d_in, const int* in_sizes, int n_in,
                              void* d_out, int out_size, void* d_ws, size_t ws_size,
                              hipStream_t stream)
{
    (void)in_sizes; (void)n_in; (void)out_size; (void)ws_size;

    const float* x = (const float*)d_in[0];
    const float* bw[8];
    for (int i = 0; i < 8; ++i) bw[i] = (const float*)d_in[1 + i];
    const float *cw[6], *cb[6], *xw[6], *xb[6];
    for (int i = 0; i < 6; ++i) {
        cw[i] = (const float*)d_in[9 + i];
        cb[i] = (const float*)d_in[15 + i];
        xw[i] = (const float*)d_in[21 + i];
        xb[i] = (const float*)d_in[27 + i];
    }

    float* ws = (float*)d_ws;
    // activation layout (floats). A0 region is reused for head outputs.
    float* A0 = ws;                           // 8*64*256*256  = 33554432
    float* A1 = ws + 33554432LL;              // 8*128*128*128 = 16777216
    float* A2 = ws + 50331648LL;              // 8*256*64*64   =  8388608
    float* A3 = ws + 58720256LL;              // 8*512*32*32   =  4194304
    float* A4 = ws + 62914560LL;              // 8*512*16*16   =  1048576
    float* A5 = ws + 63963136LL;              // 8*256*8*8     =   131072
    float* A6 = ws + 64094208LL;              // 8*256*4*4     =    32768
    float* A7 = ws + 64126976LL;              // 8*256*2*2     =     8192

    // ---- backbone: 8 stride-2 conv3x3 + ReLU ----
    launch_conv(x,  bw[0], nullptr, A0,   3, 512, 512,  64, 256, 256, 2, 1, stream);
    launch_conv(A0, bw[1], nullptr, A1,  64, 256, 256, 128, 128, 128, 2, 1, stream);
    launch_conv(A1, bw[2], nullptr, A2, 128, 128, 128, 256,  64,  64, 2, 1, stream);
    launch_conv(A2, bw[3], nullptr, A3, 256,  64,  64, 512,  32,  32, 2, 1, stream);
    launch_conv(A3, bw[4], nullptr, A4, 512,  32,  32, 512,  16,  16, 2, 1, stream);
    launch_conv(A4, bw[5], nullptr, A5, 512,  16,  16, 256,   8,   8, 2, 1, stream);
    launch_conv(A5, bw[6], nullptr, A6, 256,   8,   8, 256,   4,   4, 2, 1, stream);
    launch_conv(A6, bw[7], nullptr, A7, 256,   4,   4, 256,   2,   2, 2, 1, stream);

    // ---- heads: stride-1 conv3x3 + bias (no relu) ----
    const float* feat[6] = {A2, A3, A4, A5, A6, A7};
    const int featC[6]   = {256, 512, 512, 256, 256, 256};
    const int featF[6]   = {64, 32, 16, 8, 4, 2};
    const int clsC[6]    = {84, 126, 126, 126, 84, 84};   // na*(C+1)
    const int boxC[6]    = {16, 24, 24, 24, 16, 16};      // na*4
    const long long clsOff[6] = {0LL, 2752512LL, 3784704LL, 4042752LL, 4107264LL, 4118016LL};
    const long long boxOff[6] = {4120704LL, 4644992LL, 4841600LL, 4890752LL, 4903040LL, 4905088LL};

    for (int s = 0; s < 6; ++s) {
        int f = featF[s];
        launch_conv(feat[s], cw[s], cb[s], ws + clsOff[s],
                    featC[s], f, f, clsC[s], f, f, 1, 0, stream);
        launch_conv(feat[s], xw[s], xb[s], ws + boxOff[s],
                    featC[s], f, f, boxC[s], f, f, 1, 0, stream);
    }

    // ---- post: softmax + decode + scatter into (ids | scores | bboxes) ----
    float* o_ids = (float*)d_out;
    float* o_sc  = o_ids + (long long)BATCH * NCLASS * NANCH;
    float* o_bb  = o_sc  + (long long)BATCH * NCLASS * NANCH;
    int total = BATCH * NANCH;
    ssd_post<<<(total + 255) / 256, 256, 0, stream>>>(ws, o_ids, o_sc, o_bb);
}